// Router_28741921145440
// MI455X (gfx1250) — compile-verified
//
#include <hip/hip_runtime.h>
#include <math.h>

typedef float f4  __attribute__((ext_vector_type(4)));
typedef float v2f __attribute__((ext_vector_type(2)));
typedef float v8f __attribute__((ext_vector_type(8)));

#define N_PATCH 32768
#define C_DIM   32
#define HW      64
#define NEXP    16

// ---------------------------------------------------------------------------
// Phase A: mean-pool over H*W (64 contiguous floats per (n,c)) + L2 normalize.
// One wave32 per patch; lane == channel. 16x b128 NT loads per lane (stream,
// data read once, 256MB > 192MB L2 so keep it out of the cache).
// ---------------------------------------------------------------------------
__global__ __launch_bounds__(256) void embed_kernel(const float* __restrict__ patch,
                                                    float* __restrict__ emb) {
    const int lane = threadIdx.x & 31;
    const int wave = threadIdx.x >> 5;
    const int p    = blockIdx.x * 8 + wave;           // one patch per wave

    const f4* src = (const f4*)(patch + (size_t)p * (C_DIM * HW) + (size_t)lane * HW);
    float s = 0.0f;
#pragma unroll
    for (int i = 0; i < 16; ++i) {
        f4 v = __builtin_nontemporal_load(src + i);   // global_load_b128 TH_NT
        s += (v.x + v.y) + (v.z + v.w);
    }
    float mean = s * (1.0f / 64.0f);

    // wave-wide sum of squares (wave32)
    float sq = mean * mean;
#pragma unroll
    for (int off = 16; off >= 1; off >>= 1)
        sq += __shfl_xor(sq, off, 32);

    float nrm = fmaxf(sqrtf(sq), 1e-12f);
    emb[(size_t)p * C_DIM + lane] = mean / nrm;
}

// ---------------------------------------------------------------------------
// Phase B: logits = emb(16x32) @ keys^T(32x16) via V_WMMA_F32_16X16X4_F32,
// then per-row softmax / adaptive-threshold / mask epilogue.
// One wave32 per 16-patch tile, 8 waves per block.
// ---------------------------------------------------------------------------
__global__ __launch_bounds__(256) void router_kernel(const float* __restrict__ emb,
                                                     const float* __restrict__ keys,
                                                     const float* __restrict__ lt,
                                                     const float* __restrict__ mb,
                                                     const float* __restrict__ th,
                                                     float* __restrict__ out) {
    __shared__ float lgt[8][16 * 17];                 // per-wave 16x16 tile, pad 17

    const int lane = threadIdx.x & 31;
    const int wave = threadIdx.x >> 5;
    const int tile = blockIdx.x * 8 + wave;           // 16 patches per tile
    const int half = lane >> 4;                       // 0: lanes 0-15, 1: 16-31
    const int l16  = lane & 15;

    // --- WMMA accumulation over K=32 in 8 steps of K=4 -----------------------
    // A (16x4 f32): lanes 0-15 hold (K=kb,kb+1), lanes 16-31 hold (K=kb+2,kb+3),
    // M = lane&15.  B (4x16): lanes hold N=lane&15, same K split per half-wave.
    v8f acc = {};
    const int arow = tile * 16 + l16;
#pragma unroll
    for (int s = 0; s < 8; ++s) {
        const int kb = s * 4 + half * 2;
        v2f a = *(const v2f*)(emb  + (size_t)arow * 32 + kb);   // A[M][kb..kb+1]
        v2f b = *(const v2f*)(keys + (size_t)l16  * 32 + kb);   // keys[N][kb..kb+1] = B[kb..][N]
        acc = __builtin_amdgcn_wmma_f32_16x16x4_f32(
            /*neg_a=*/false, a, /*neg_b=*/false, b,
            /*c_mod=*/(short)0, acc, /*reuse_a=*/false, /*reuse_b=*/false);
    }

    // C layout: VGPR r -> row (r + half*8), col l16. Spill tile through LDS so
    // each of lanes 0-15 owns one full row (one patch).
#pragma unroll
    for (int r = 0; r < 8; ++r)
        lgt[wave][(r + half * 8) * 17 + l16] = acc[r];
    __syncthreads();

    if (lane < 16) {
        const int   row   = tile * 16 + lane;
        const float temp  = lt[0];
        const float beta  = mb[0] + 1e-8f;
        const float bthr  = th[0];

        // ---- softmax (stable) ----
        float w[16];
        float mx = -1e30f;
#pragma unroll
        for (int j = 0; j < 16; ++j) {
            float v = lgt[wave][lane * 17 + j] / temp + 1e-8f;
            w[j] = v;
            mx = fmaxf(mx, v);
        }
        float sume = 0.0f;
#pragma unroll
        for (int j = 0; j < 16; ++j) { w[j] = __expf(w[j] - mx); sume += w[j]; }
        const float isum = 1.0f / sume;
#pragma unroll
        for (int j = 0; j < 16; ++j) w[j] *= isum;

        // ---- top-5 (only sorted[0..4] are consumed by the reference) ----
        float work[16];
#pragma unroll
        for (int j = 0; j < 16; ++j) work[j] = w[j];
        float top[5];
#pragma unroll
        for (int p = 0; p < 5; ++p) {
            float best = -1e30f; int bj = 0;
#pragma unroll
            for (int j = 0; j < 16; ++j)
                if (work[j] > best) { best = work[j]; bj = j; }
            top[p] = best;
#pragma unroll
            for (int j = 0; j < 16; ++j)
                work[j] = (j == bj) ? -1e30f : work[j];
        }

        const float s0        = top[0];
        const float mean_rest = 0.25f * (top[1] + top[2] + top[3] + top[4]);
        float gap = (s0 - mean_rest) / (s0 + 1e-8f);
        gap = fminf(fmaxf(gap, 0.0f), 1.0f);

        float sum = 0.0f;
#pragma unroll
        for (int j = 0; j < 16; ++j) sum += w[j];
        const float mean = sum * (1.0f / 16.0f);
        float var = 0.0f;
#pragma unroll
        for (int j = 0; j < 16; ++j) { float d = w[j] - mean; var += d * d; }
        const float stdw = sqrtf(var * (1.0f / 15.0f));     // ddof=1

        float ent = 0.0f;
#pragma unroll
        for (int j = 0; j < 16; ++j) ent -= w[j] * __logf(w[j] + 1e-18f);
        const float ent_c = 1.0f - ent * (1.0f / 2.7725887222397811f); // /log(16)

        const float max_c  = 1.0f - s0;
        const float factor = 0.4f * max_c + 0.3f * ent_c + 0.3f * gap;
        float adap = bthr * (0.5f + factor);
        const float min_thr = fmaxf(0.05f, mean - 0.5f * stdw);
        const float max_thr = fminf(0.7f, s0 - 0.1f * stdw);
        adap = fminf(fmaxf(adap, min_thr), max_thr);
        adap = fminf(adap, top[3] * 0.9f);                  // kth (4th largest) * 0.9

        // ---- sigmoid mask + renormalize ----
        float fw[16];
        float fsum = 0.0f;
#pragma unroll
        for (int j = 0; j < 16; ++j) {
            float m = 1.0f / (1.0f + __expf(-beta * (w[j] - adap)));
            fw[j] = w[j] * m;
            fsum += fw[j];
        }
        const float rs = 1.0f / fmaxf(fsum, 1e-8f);

        f4* o = (f4*)(out + (size_t)row * 16);
#pragma unroll
        for (int q = 0; q < 4; ++q) {
            f4 v = { fw[q * 4 + 0] * rs, fw[q * 4 + 1] * rs,
                     fw[q * 4 + 2] * rs, fw[q * 4 + 3] * rs };
            o[q] = v;
        }
    }
}

// ---------------------------------------------------------------------------
extern "C" void kernel_launch(void* const* d_in, const int* in_sizes, int n_in,
                              void* d_out, int out_size, void* d_ws, size_t ws_size,
                              hipStream_t stream) {
    const float* patch = (const float*)d_in[0];   // (32768, 32, 8, 8) f32
    const float* keys  = (const float*)d_in[1];   // (16, 32) f32
    const float* lt    = (const float*)d_in[2];   // scalar
    const float* mb    = (const float*)d_in[3];   // scalar
    const float* th    = (const float*)d_in[4];   // scalar
    float*       out   = (float*)d_out;           // (32768, 16) f32
    float*       emb   = (float*)d_ws;            // (32768, 32) f32 = 4 MB scratch

    // Phase A: 1 wave per patch, 8 waves per 256-thread block.
    embed_kernel<<<N_PATCH / 8, 256, 0, stream>>>(patch, emb);

    // Phase B: 1 wave per 16-patch tile, 8 tiles per block -> 256 blocks.
    router_kernel<<<(N_PATCH / 16) / 8, 256, 0, stream>>>(emb, keys, lt, mb, th, out);
}